// MultiStepSpikeFFN_14585708937548
// MI455X (gfx1250) — compile-verified
//
#include <hip/hip_runtime.h>

// MultiStepSpikeFFN for MI455X (gfx1250, wave32).
// Pipeline: GEMM1(fp32 WMMA) -> LIF scan (in place) -> GEMM2(fp32 WMMA)
//           -> LIF scan (in place) -> residual + LayerNorm.
// Scratch: Y1 (M*H f32, 64MB) + Y2 (M*D f32, 32MB) in d_ws.

typedef float v2f __attribute__((ext_vector_type(2)));
typedef float v8f __attribute__((ext_vector_type(8)));

#define BSTRIDE 80  // pair-row stride in float2: 160 dwords = 32 banks -> lane halves conflict-free

// Y[M,N] = A[M,K] @ W^T + bias, where W is [N,K] row-major (so B[k][n] = W[n*K+k]).
// grid = (M/256, N/64), block = 256 threads (8 waves), wave -> 32x64 output tile
// (2 M-fragments x 4 N-fragments = 8 f32 WMMA accumulators).
// B is staged in LDS pre-swizzled as K-pairs so each lane's WMMA B-fragment is a
// single aligned ds_load_b64 straight into an even VGPR pair; fragments for the
// next K-step are prefetched (reg double-buffer) so the LDS latency hides under
// the current step's 8 WMMAs instead of stalling on s_wait_dscnt 0.
template <int K>
__global__ void __launch_bounds__(256)
gemm_bias_wmma(const float* __restrict__ A, const float* __restrict__ W,
               const float* __restrict__ bias, float* __restrict__ Y, int N)
{
    constexpr int KP = K / 2;                 // number of K-pairs
    __shared__ v2f Bp[KP * BSTRIDE];          // Bp[kp*BSTRIDE + n] = {B[2kp][n], B[2kp+1][n]}

    const int tid   = threadIdx.x;
    const int lane  = tid & 31;
    const int wave  = tid >> 5;
    const int nl    = lane & 15;              // N (or M-row) selector within fragment
    const int half  = lane >> 4;              // 0: K=k0,k0+1 ; 1: K=k0+2,k0+3
    const int n0    = blockIdx.y * 64;
    const int m0    = blockIdx.x * 256 + wave * 32;

    // Stage B slice: consecutive tids read consecutive K-pairs (contiguous 8B in W).
    for (int idx = tid; idx < KP * 64; idx += 256) {
        int kp = idx % KP;
        int n  = idx / KP;
        Bp[kp * BSTRIDE + n] = *(const v2f*)&W[(size_t)(n0 + n) * K + 2 * kp];
    }
    __syncthreads();

    // A fragments: lane nl holds rows m0+nl and m0+16+nl, K pair starting at 2*half.
    const float* a0p = A + (size_t)(m0 + nl) * K + 2 * half;
    const float* a1p = a0p + (size_t)16 * K;

    // Preload B fragments for k0 = 0.
    const v2f* bp0 = &Bp[half * BSTRIDE + nl];
    v2f b0 = bp0[0], b1 = bp0[16], b2 = bp0[32], b3 = bp0[48];

    v8f c00 = {}, c01 = {}, c02 = {}, c03 = {};
    v8f c10 = {}, c11 = {}, c12 = {}, c13 = {};
    #pragma unroll 4
    for (int k0 = 0; k0 < K; k0 += 4) {
        v2f a0 = *(const v2f*)(a0p + k0);
        v2f a1 = *(const v2f*)(a1p + k0);
        // Prefetch next K-step's B fragments (wraps on the final step; redundant
        // but keeps the trip count divisible by the unroll factor).
        const int kn = (k0 + 4) & (K - 1);
        const v2f* bp = &Bp[(kn / 2 + half) * BSTRIDE + nl];
        v2f nb0 = bp[0];
        v2f nb1 = bp[16];
        v2f nb2 = bp[32];
        v2f nb3 = bp[48];
        c00 = __builtin_amdgcn_wmma_f32_16x16x4_f32(false, a0, false, b0, (short)0, c00, false, false);
        c01 = __builtin_amdgcn_wmma_f32_16x16x4_f32(false, a0, false, b1, (short)0, c01, false, false);
        c02 = __builtin_amdgcn_wmma_f32_16x16x4_f32(false, a0, false, b2, (short)0, c02, false, false);
        c03 = __builtin_amdgcn_wmma_f32_16x16x4_f32(false, a0, false, b3, (short)0, c03, false, false);
        c10 = __builtin_amdgcn_wmma_f32_16x16x4_f32(false, a1, false, b0, (short)0, c10, false, false);
        c11 = __builtin_amdgcn_wmma_f32_16x16x4_f32(false, a1, false, b1, (short)0, c11, false, false);
        c12 = __builtin_amdgcn_wmma_f32_16x16x4_f32(false, a1, false, b2, (short)0, c12, false, false);
        c13 = __builtin_amdgcn_wmma_f32_16x16x4_f32(false, a1, false, b3, (short)0, c13, false, false);
        b0 = nb0; b1 = nb1; b2 = nb2; b3 = nb3;
    }

    // C/D layout: VGPR i -> row base+i (lanes 0-15) / base+8+i (lanes 16-31), col = tile + nl.
    const float bv0 = bias[n0 + nl];
    const float bv1 = bias[n0 + 16 + nl];
    const float bv2 = bias[n0 + 32 + nl];
    const float bv3 = bias[n0 + 48 + nl];
    const int rowBase = m0 + (half << 3);
    #pragma unroll
    for (int i = 0; i < 8; ++i) {
        size_t off = (size_t)(rowBase + i) * N + n0 + nl;
        Y[off]      = c00[i] + bv0;
        Y[off + 16] = c01[i] + bv1;
        Y[off + 32] = c02[i] + bv2;
        Y[off + 48] = c03[i] + bv3;
        size_t off2 = off + (size_t)16 * N;
        Y[off2]      = c10[i] + bv0;
        Y[off2 + 16] = c11[i] + bv1;
        Y[off2 + 32] = c12[i] + bv2;
        Y[off2 + 48] = c13[i] + bv3;
    }
}

// In-place multi-step LIF over Y[B, T, C]: one thread per (b, c) chain.
// h = (v + x)/2 ; s = (h >= 1) ; v = s ? 0 : h.  8-timestep batches keep 8
// coalesced loads in flight instead of a pure load-use latency chain.
template <int C>
__global__ void __launch_bounds__(256)
lif_scan_inplace(float* __restrict__ Y, int T)
{
    const int idx = blockIdx.x * blockDim.x + threadIdx.x;
    const int b = idx / C;
    const int c = idx % C;
    float* p = Y + (size_t)b * T * C + c;

    float v = 0.0f;
    for (int t = 0; t < T; t += 8) {
        float xv[8];
        #pragma unroll
        for (int i = 0; i < 8; ++i) xv[i] = p[(size_t)(t + i) * C];
        #pragma unroll
        for (int i = 0; i < 8; ++i) {
            float h = 0.5f * (v + xv[i]);
            bool fire = (h >= 1.0f);
            xv[i] = fire ? 1.0f : 0.0f;
            v = fire ? 0.0f : h;
        }
        #pragma unroll
        for (int i = 0; i < 8; ++i) p[(size_t)(t + i) * C] = xv[i];
    }
}

// out[row,:] = LayerNorm(x[row,:] + s[row,:]) * w + b, row of 128, one wave32 per row.
__global__ void __launch_bounds__(256)
resid_layernorm(const float* __restrict__ x, const float* __restrict__ s,
                const float* __restrict__ w, const float* __restrict__ b,
                float* __restrict__ out)
{
    const int row  = blockIdx.x * 8 + (threadIdx.x >> 5);
    const int lane = threadIdx.x & 31;
    const size_t base = (size_t)row * 128 + lane * 4;

    const float4 xv = *(const float4*)(x + base);
    const float4 sv = *(const float4*)(s + base);
    float y0 = xv.x + sv.x, y1 = xv.y + sv.y, y2 = xv.z + sv.z, y3 = xv.w + sv.w;

    float sum = y0 + y1 + y2 + y3;
    #pragma unroll
    for (int m = 16; m; m >>= 1) sum += __shfl_xor(sum, m, 32);
    const float mu = sum * (1.0f / 128.0f);

    float d0 = y0 - mu, d1 = y1 - mu, d2 = y2 - mu, d3 = y3 - mu;
    float ss = d0 * d0 + d1 * d1 + d2 * d2 + d3 * d3;
    #pragma unroll
    for (int m = 16; m; m >>= 1) ss += __shfl_xor(ss, m, 32);
    const float r = rsqrtf(ss * (1.0f / 128.0f) + 1e-5f);

    const float4 wv = *(const float4*)(w + lane * 4);
    const float4 bv = *(const float4*)(b + lane * 4);
    float4 o;
    o.x = d0 * r * wv.x + bv.x;
    o.y = d1 * r * wv.y + bv.y;
    o.z = d2 * r * wv.z + bv.z;
    o.w = d3 * r * wv.w + bv.w;
    *(float4*)(out + base) = o;
}

extern "C" void kernel_launch(void* const* d_in, const int* in_sizes, int n_in,
                              void* d_out, int out_size, void* d_ws, size_t ws_size,
                              hipStream_t stream)
{
    const float* x   = (const float*)d_in[0];
    const float* W1  = (const float*)d_in[1];  // [H, D]
    const float* b1  = (const float*)d_in[2];  // [H]
    const float* W2  = (const float*)d_in[3];  // [D, H]
    const float* b2  = (const float*)d_in[4];  // [D]
    const float* lnw = (const float*)d_in[5];  // [D]
    const float* lnb = (const float*)d_in[6];  // [D]
    float* out = (float*)d_out;

    const int B = 32, T = 2048, D = 128, H = 256;
    const int M = B * T;  // 65536 independent rows

    float* Y1 = (float*)d_ws;            // [M, H] fp32 (spikes written in place)
    float* Y2 = Y1 + (size_t)M * H;      // [M, D] fp32 (spikes written in place)

    // h = x @ W1^T + b1
    gemm_bias_wmma<128><<<dim3(M / 256, H / 64), 256, 0, stream>>>(x, W1, b1, Y1, H);
    // first LIF scan (in place): 32*256 = 8192 chains
    lif_scan_inplace<256><<<(B * H) / 256, 256, 0, stream>>>(Y1, T);
    // h = spikes @ W2^T + b2
    gemm_bias_wmma<256><<<dim3(M / 256, D / 64), 256, 0, stream>>>(Y1, W2, b2, Y2, D);
    // second LIF scan (in place): 32*128 = 4096 chains
    lif_scan_inplace<128><<<(B * D) / 256, 256, 0, stream>>>(Y2, T);
    // residual + LayerNorm
    resid_layernorm<<<M / 8, 256, 0, stream>>>(x, Y2, lnw, lnb, out);
}